// MillionsMOE_4269197492236
// MI455X (gfx1250) — compile-verified
//
#include <hip/hip_runtime.h>
#include <hip/hip_bf16.h>

// PKM-MoE forward for MI455X (gfx1250, wave32).
// Roofline: gather of w_down/w_up rows = 536 MB -> ~23us @ 23.3 TB/s is the
// floor; GEMMs (3.2 GFLOP) run on v_wmma_f32_16x16x4_f32 so they stay exact
// fp32 (top-k ordering identical to reference) and far off the critical path.

typedef float v2f __attribute__((ext_vector_type(2)));
typedef float v8f __attribute__((ext_vector_type(8)));

#define N_HEADS 8
#define D_KEYS 128
#define HALF_K 64
#define N_KEYS 256
#define TOPK 8
#define D_MODEL 512
#define QDIM (N_HEADS * D_KEYS) // 1024
#define NEG_INF (-3.4e38f)

// ---------------------------------------------------------------------------
// Kernel 1: q = x @ Wq + bq.   x:(B,512)  Wq:(512,1024)  q:(B,1024)
// One 16x16 output tile per wave; 4 waves/block cover 16 rows x 64 cols.
// WMMA f32 16x16x4 fragment layouts (ISA 7.12.2, 32-bit):
//   A[16x4]: lane L holds row M=L%16, VGPRs {0,1} hold K = 2*(L/16)+{0,1}
//   B[4x16]: lane L holds col N=L%16, VGPRs {0,1} hold K = 2*(L/16)+{0,1}
//   C/D    : VGPR g -> M = g + 8*(L/16), N = L%16
// ---------------------------------------------------------------------------
__global__ __launch_bounds__(128) void pkm_qproj(
    const float* __restrict__ x, const float* __restrict__ Wq,
    const float* __restrict__ bq, float* __restrict__ q) {
  const int lane = threadIdx.x & 31;
  const int wave = threadIdx.x >> 5;
  const int lm   = lane & 15;
  const int hi   = lane >> 4;          // which K-pair this lane owns
  const int m0   = blockIdx.x * 16;
  const int n0   = blockIdx.y * 64 + wave * 16;

  const float* arow = x + (size_t)(m0 + lm) * D_MODEL + 2 * hi;
  const float* bcol = Wq + n0 + lm;

  v8f c = {};
  for (int k = 0; k < D_MODEL; k += 4) {
    v2f a = *(const v2f*)(arow + k);                 // K = k+2*hi, k+2*hi+1
    v2f b;
    b.x = bcol[(size_t)(k + 2 * hi + 0) * QDIM];
    b.y = bcol[(size_t)(k + 2 * hi + 1) * QDIM];
    c = __builtin_amdgcn_wmma_f32_16x16x4_f32(false, a, false, b, (short)0, c,
                                              false, false);
  }
  const float bias = bq[n0 + lm];
#pragma unroll
  for (int g = 0; g < 8; ++g) {
    const int row = m0 + g + 8 * hi;
    q[(size_t)row * QDIM + n0 + lm] = c[g] + bias;
  }
}

// ---------------------------------------------------------------------------
// Kernel 2: per (16-token tile, head): s1 = q1 @ K1^T, s2 = q2 @ K2^T via
// WMMA into LDS, then stable top-8 + 8x8 combine + top-8 + softmax.
// Block = 256 threads (8 waves). 32 wmma n-tile jobs (16 per score half).
// ---------------------------------------------------------------------------
__global__ __launch_bounds__(256) void pkm_scores_topk(
    const float* __restrict__ q, const float* __restrict__ keys,
    float* __restrict__ gates, int* __restrict__ topidx) {
  __shared__ float sS[2][16][N_KEYS + 1];  // +1 pad: kill bank conflicts
  const int lane = threadIdx.x & 31;
  const int wave = threadIdx.x >> 5;
  const int lm   = lane & 15;
  const int hi   = lane >> 4;
  const int b0   = blockIdx.x * 16;
  const int h    = blockIdx.y;

  for (int job = wave; job < 32; job += 8) {
    const int ssel = job >> 4;           // 0 -> s1, 1 -> s2
    const int n0   = (job & 15) * 16;    // key tile
    const float* arow =
        q + (size_t)(b0 + lm) * QDIM + h * D_KEYS + ssel * HALF_K + 2 * hi;
    // keys[h][ssel][n][k] row-major; B frag wants B[k][n] = keys[...][n][k]
    const float* krow =
        keys + ((size_t)(h * 2 + ssel) * N_KEYS + (n0 + lm)) * HALF_K + 2 * hi;
    v8f c = {};
    for (int k = 0; k < HALF_K; k += 4) {
      v2f a = *(const v2f*)(arow + k);
      v2f b = *(const v2f*)(krow + k);
      c = __builtin_amdgcn_wmma_f32_16x16x4_f32(false, a, false, b, (short)0,
                                                c, false, false);
    }
#pragma unroll
    for (int g = 0; g < 8; ++g) sS[ssel][g + 8 * hi][n0 + lm] = c[g];
  }
  __syncthreads();

  if (threadIdx.x < 16) {
    const int t = threadIdx.x;
    float tv1[TOPK], tv2[TOPK];
    int   ti1[TOPK], ti2[TOPK];
#pragma unroll
    for (int s = 0; s < TOPK; ++s) {
      tv1[s] = NEG_INF; tv2[s] = NEG_INF; ti1[s] = 0; ti2[s] = 0;
    }
    // Stable top-8 per half: ascending scan + strict-> insertion keeps the
    // lowest index first on ties, matching lax.top_k.
    for (int n = 0; n < N_KEYS; ++n) {
      float v = sS[0][t][n]; int i = n;
#pragma unroll
      for (int s = 0; s < TOPK; ++s)
        if (v > tv1[s]) {
          float fv = tv1[s]; tv1[s] = v; v = fv;
          int   fi = ti1[s]; ti1[s] = i; i = fi;
        }
      v = sS[1][t][n]; i = n;
#pragma unroll
      for (int s = 0; s < TOPK; ++s)
        if (v > tv2[s]) {
          float fv = tv2[s]; tv2[s] = v; v = fv;
          int   fi = ti2[s]; ti2[s] = i; i = fi;
        }
    }
    // Combine 8x8 candidate sums (flattened i-major, ascending => stable).
    float cv[TOPK]; int ci[TOPK];
#pragma unroll
    for (int s = 0; s < TOPK; ++s) { cv[s] = NEG_INF; ci[s] = 0; }
    for (int i = 0; i < TOPK; ++i)
      for (int j = 0; j < TOPK; ++j) {
        float v = tv1[i] + tv2[j];
        int   ii = ti1[i] * N_KEYS + ti2[j];
#pragma unroll
        for (int s = 0; s < TOPK; ++s)
          if (v > cv[s]) {
            float fv = cv[s]; cv[s] = v; v = fv;
            int   fi = ci[s]; ci[s] = ii; ii = fi;
          }
      }
    // Softmax over the 8 selected scores (cv[0] is the max: list is sorted).
    float e[TOPK], sum = 0.f;
#pragma unroll
    for (int s = 0; s < TOPK; ++s) { e[s] = __expf(cv[s] - cv[0]); sum += e[s]; }
    const float inv = 1.f / sum;
    const size_t base = ((size_t)(b0 + t) * N_HEADS + h) * TOPK;
#pragma unroll
    for (int s = 0; s < TOPK; ++s) {
      gates[base + s]  = e[s] * inv;
      topidx[base + s] = ci[s];
    }
  }
}

// ---------------------------------------------------------------------------
// Kernel 3 (HBM-bound): one block per token. Reads 64 w_down rows (inner
// products, stride-32 coalesced + wave32 shfl_xor reduce) and 64 w_up rows
// (coalesced column accumulation). Prefetches w_up lines while w_down inners
// are in flight (global_prefetch_b8).
// ---------------------------------------------------------------------------
__global__ __launch_bounds__(256) void pkm_gather_out(
    const float* __restrict__ x, const float* __restrict__ w_down,
    const float* __restrict__ w_up, const float* __restrict__ gates,
    const int* __restrict__ topidx, float* __restrict__ out) {
  __shared__ float xs[D_MODEL];
  __shared__ int   sIdx[64];
  __shared__ float sGate[64];
  __shared__ float sVal[64];

  const int b    = blockIdx.x;
  const int t    = threadIdx.x;
  const int lane = t & 31;
  const int wave = t >> 5;

  if (t < 64) {
    sIdx[t]  = topidx[(size_t)b * 64 + t];
    sGate[t] = gates[(size_t)b * 64 + t];
  }
  xs[t]       = x[(size_t)b * D_MODEL + t];
  xs[t + 256] = x[(size_t)b * D_MODEL + t + 256];
  __syncthreads();

  // Prefetch all 64 w_up rows (16 x 128B lines each => 1024 lines, 4/thread).
#pragma unroll
  for (int i = 0; i < 4; ++i) {
    const int e = t + i * 256;
    const char* p = (const char*)(w_up + (size_t)sIdx[e >> 4] * D_MODEL) +
                    (e & 15) * 128;
    __builtin_prefetch(p, 0, 0);
  }

  // Inner products: wave w owns rows [8w, 8w+8).
  for (int r = wave * 8; r < wave * 8 + 8; ++r) {
    const float* wd = w_down + (size_t)sIdx[r] * D_MODEL;
    float p = 0.f;
#pragma unroll
    for (int j = 0; j < D_MODEL; j += 32) p = fmaf(xs[j + lane], wd[j + lane], p);
#pragma unroll
    for (int off = 16; off > 0; off >>= 1) p += __shfl_xor(p, off, 32);
    if (lane == 0) sVal[r] = fmaxf(p, 0.f) * sGate[r];
  }
  __syncthreads();

  // out[b,:] = sum_r val[r] * w_up[idx[r],:]  -- each thread owns 2 dims.
  float acc0 = 0.f, acc1 = 0.f;
  for (int r = 0; r < 64; ++r) {
    const float* wu = w_up + (size_t)sIdx[r] * D_MODEL;
    const float v = sVal[r];
    acc0 = fmaf(v, wu[t], acc0);
    acc1 = fmaf(v, wu[t + 256], acc1);
  }
  out[(size_t)b * D_MODEL + t]       = acc0;
  out[(size_t)b * D_MODEL + t + 256] = acc1;
}

// ---------------------------------------------------------------------------
extern "C" void kernel_launch(void* const* d_in, const int* in_sizes, int n_in,
                              void* d_out, int out_size, void* d_ws,
                              size_t ws_size, hipStream_t stream) {
  const float* x      = (const float*)d_in[0];  // (N,T,512) -> (B,512)
  const float* Wq     = (const float*)d_in[1];  // (512,1024)
  const float* bq     = (const float*)d_in[2];  // (1024)
  const float* keys   = (const float*)d_in[3];  // (8,2,256,64)
  const float* w_down = (const float*)d_in[4];  // (65536,512)
  const float* w_up   = (const float*)d_in[5];  // (65536,512)
  float* out          = (float*)d_out;          // (B,512)

  const int B = in_sizes[0] / D_MODEL;          // 2048

  // Workspace: q (B*1024 f32) | gates (B*64 f32) | topidx (B*64 i32)
  float* q      = (float*)d_ws;
  float* gates  = q + (size_t)B * QDIM;
  int*   topidx = (int*)(gates + (size_t)B * 64);

  pkm_qproj<<<dim3(B / 16, QDIM / 64), 128, 0, stream>>>(x, Wq, bq, q);
  pkm_scores_topk<<<dim3(B / 16, N_HEADS), 256, 0, stream>>>(q, keys, gates,
                                                             topidx);
  pkm_gather_out<<<B, 256, 0, stream>>>(x, w_down, w_up, gates, topidx, out);
}